// GMSA_27865747816607
// MI455X (gfx1250) — compile-verified
//
#include <hip/hip_runtime.h>
#include <hip/hip_bf16.h>

typedef float v2f __attribute__((ext_vector_type(2)));
typedef float v8f __attribute__((ext_vector_type(8)));

// ---------------------------------------------------------------------------
// fp32 WMMA GEMM:  C(M,N) = A(M,K) * B(K,N) [+ bias(N)]
//
//   Block = 256 threads = 8 waves. Block computes a 128x64 C macro-tile:
//   wave w owns rows [bM*128 + w*16, +16), all 64 columns (4 accumulators).
//   B K-chunks (16 x 64) are staged once per block into LDS with
//   global_load_async_to_lds_b128 (ASYNCcnt), then every wave reads its
//   fragments from LDS -> 8x reuse of B, 4 WMMAs per A fragment load.
//
//   M % 128 == 0 and K % 4 == 0 hold for every call site here; N edges are
//   handled by predicated staging + masked stores.
//   TRANSB: B stored (N x K) row-major; logical B[k][n] = Bst[n*ldb + k].
// ---------------------------------------------------------------------------
#define SB0 68   // LDS row stride (floats) for TRANSB=0 layout [k][n]
#define SB1 20   // LDS col stride (floats) for TRANSB=1 layout [n][k]

template <int TRANSB>
__global__ void __launch_bounds__(256)
gemm_f32_wmma(const float* __restrict__ A, const float* __restrict__ B,
              const float* __restrict__ bias, float* __restrict__ C,
              int M, int N, int K, int lda, int ldb, int ldc)
{
    __shared__ float bs[1280];          // max(16*SB0, 64*SB1)

    const int tilesN = (N + 63) >> 6;
    const int bM = blockIdx.x / tilesN;
    const int bN = blockIdx.x % tilesN;
    const int n0 = bN << 6;

    const int lane = threadIdx.x & 31;
    const int wave = threadIdx.x >> 5;
    const int lm   = lane & 15;         // row (A) / col (B,C) within 16-tile
    const int kh   = lane >> 4;         // K-half select (ISA frag layout)

    const int m = bM * 128 + wave * 16 + lm;            // A row of this lane
    const float* __restrict__ Ap = A + (size_t)m * lda;

    v8f acc[4] = {{}, {}, {}, {}};

    for (int kb = 0; kb < K; kb += 16) {
        const int chunk = (K - kb < 16) ? (K - kb) : 16;

        // ---- stage B chunk into LDS (async, per-lane 16B copies) ----------
        if (TRANSB == 0) {
            const int r  = threadIdx.x >> 4;        // k-row in chunk: 0..15
            const int cq = (threadIdx.x & 15) << 2; // column quad: 0,4,..,60
            if (r < chunk) {
                const int nb  = n0 + cq;
                const int rem = N - nb;
                const float* gp = B + (size_t)(kb + r) * ldb + nb;
                unsigned lds = (unsigned)(size_t)(void*)&bs[r * SB0 + cq];
                if (rem >= 4) {
                    asm volatile("global_load_async_to_lds_b128 %0, %1, off"
                                 :: "v"(lds), "v"((unsigned long long)gp)
                                 : "memory");
                } else if (rem > 0) {
                    for (int i = 0; i < rem; ++i)
                        asm volatile("global_load_async_to_lds_b32 %0, %1, off"
                                     :: "v"(lds + 4u * i),
                                        "v"((unsigned long long)(gp + i))
                                     : "memory");
                }
            }
        } else {
            const int c  = threadIdx.x >> 2;        // column: 0..63
            const int kq = (threadIdx.x & 3) << 2;  // k quad: 0,4,8,12
            if (kq < chunk && (n0 + c) < N) {
                const float* gp = B + (size_t)(n0 + c) * ldb + kb + kq;
                unsigned lds = (unsigned)(size_t)(void*)&bs[c * SB1 + kq];
                asm volatile("global_load_async_to_lds_b128 %0, %1, off"
                             :: "v"(lds), "v"((unsigned long long)gp)
                             : "memory");
            }
        }
        asm volatile("s_wait_asynccnt 0x0" ::: "memory");
        __syncthreads();

        // ---- compute: 1 A frag feeds 4 WMMAs --------------------------------
        const float* __restrict__ Ak = Ap + kb;
        #pragma unroll 4
        for (int ks = 0; ks < chunk; ks += 4) {
            const int k0 = ks + 2 * kh;
            v2f a;
            a.x = Ak[k0];
            a.y = Ak[k0 + 1];
            #pragma unroll
            for (int t = 0; t < 4; ++t) {
                v2f b;
                if (TRANSB == 0) {
                    b.x = bs[k0 * SB0       + 16 * t + lm];
                    b.y = bs[(k0 + 1) * SB0 + 16 * t + lm];
                } else {
                    b.x = bs[(16 * t + lm) * SB1 + k0];
                    b.y = bs[(16 * t + lm) * SB1 + k0 + 1];
                }
                acc[t] = __builtin_amdgcn_wmma_f32_16x16x4_f32(
                    false, a, false, b, (short)0, acc[t], false, false);
            }
        }
        __syncthreads();                 // protect LDS before next stage
    }

    // ---- epilogue: masked store + optional bias ----------------------------
    const int rowBase = bM * 128 + wave * 16 + 8 * kh;   // C/D frag layout
    #pragma unroll
    for (int t = 0; t < 4; ++t) {
        const int nCol = n0 + 16 * t + lm;
        if (nCol < N) {
            const float bv = bias ? bias[nCol] : 0.0f;
            float* Cp = C + (size_t)rowBase * ldc + nCol;
            #pragma unroll
            for (int j = 0; j < 8; ++j)
                Cp[(size_t)j * ldc] = acc[t][j] + bv;
        }
    }
}

// ---------------------------------------------------------------------------
// Small helper kernels
// ---------------------------------------------------------------------------
__global__ void init_zero_kernel(float* __restrict__ p, int n)
{
    for (int i = threadIdx.x; i < n; i += blockDim.x) p[i] = 0.0f;
}

// Per-channel sum / sum-of-squares over x (rows x 360), LDS-atomic partials.
__global__ void __launch_bounds__(256)
bn_stats_kernel(const float* __restrict__ x, float* __restrict__ stats,
                int rowsPerBlock)
{
    __shared__ float ssum[360];
    __shared__ float ssq[360];
    for (int i = threadIdx.x; i < 360; i += 256) { ssum[i] = 0.f; ssq[i] = 0.f; }
    __syncthreads();

    const size_t base  = (size_t)blockIdx.x * rowsPerBlock * 360;
    const int    nElem = rowsPerBlock * 360;
    for (int i = threadIdx.x; i < nElem; i += 256) {
        float val = x[base + i];
        int   ch  = i % 360;
        atomicAdd(&ssum[ch], val);          // ds_add_f32
        atomicAdd(&ssq[ch],  val * val);
    }
    __syncthreads();
    for (int i = threadIdx.x; i < 360; i += 256) {
        atomicAdd(&stats[i],        ssum[i]);
        atomicAdd(&stats[360 + i],  ssq[i]);
    }
}

__global__ void __launch_bounds__(256)
bn_apply_kernel(float* __restrict__ x, const float* __restrict__ stats,
                const float* __restrict__ gamma, const float* __restrict__ beta,
                float invCount)
{
    size_t i = (size_t)blockIdx.x * 256 + threadIdx.x;   // total = 65536*360
    int ch = (int)(i % 360);
    float mean = stats[ch] * invCount;
    float var  = stats[360 + ch] * invCount - mean * mean;
    x[i] = (x[i] - mean) * rsqrtf(var + 1e-5f) * gamma[ch] + beta[ch];
}

// Gather windows out of x (65536 x 360). Roll(-ws/2) folded into the gather:
// rolled coord j reads source coord (j + ws/2) mod 128.
__global__ void __launch_bounds__(256)
pack_qv_kernel(const float* __restrict__ x, float* __restrict__ q,
               float* __restrict__ v, int wsz, int chanOff,
               long total, int D, int g)
{
    long idx = (long)blockIdx.x * 256 + threadIdx.x;
    if (idx >= total) return;
    int n = (int)(idx / D);
    int d = (int)(idx % D);
    int wi = n % g; int t = n / g; int hi = t % g; int b = t / g;
    int ci = d % 60; int t2 = d / 60; int dw = t2 % wsz; int dh = t2 / wsz;
    int sh = hi * wsz + dh + (wsz >> 1); if (sh >= 128) sh -= 128;
    int sw = wi * wsz + dw + (wsz >> 1); if (sw >= 128) sw -= 128;
    size_t src = (((size_t)b * 128 + sh) * 128 + sw) * 360 + chanOff + ci;
    q[idx] = x[src];
    v[idx] = x[src + 60];
}

// In-place row softmax on S (gridDim.x rows of length N) living in d_out.
__global__ void __launch_bounds__(256)
softmax_rows_kernel(float* __restrict__ S, int N)
{
    __shared__ float red[256];
    const int tid = threadIdx.x;
    float* row = S + (size_t)blockIdx.x * N;

    float m = -3.402823466e38f;
    for (int i = tid; i < N; i += 256) m = fmaxf(m, row[i]);
    red[tid] = m; __syncthreads();
    for (int s = 128; s > 0; s >>= 1) {
        if (tid < s) red[tid] = fmaxf(red[tid], red[tid + s]);
        __syncthreads();
    }
    const float mx = red[0]; __syncthreads();

    float sum = 0.f;
    for (int i = tid; i < N; i += 256) {
        float e = __expf(row[i] - mx);
        row[i] = e;
        sum += e;
    }
    red[tid] = sum; __syncthreads();
    for (int s = 128; s > 0; s >>= 1) {
        if (tid < s) red[tid] += red[tid + s];
        __syncthreads();
    }
    const float inv = 1.0f / red[0];
    for (int i = tid; i < N; i += 256) row[i] *= inv;
}

// Scatter windowed Y (N x D) back into y-concat (65536 x 180). Roll(+ws/2)
// folded in: rolled coord j writes destination (j + ws/2) mod 128.
__global__ void __launch_bounds__(256)
scatter_y_kernel(const float* __restrict__ yp, float* __restrict__ ybuf,
                 int wsz, int chanOff, long total, int D, int g)
{
    long idx = (long)blockIdx.x * 256 + threadIdx.x;
    if (idx >= total) return;
    int n = (int)(idx / D);
    int d = (int)(idx % D);
    int wi = n % g; int t = n / g; int hi = t % g; int b = t / g;
    int ci = d % 60; int t2 = d / 60; int dw = t2 % wsz; int dh = t2 / wsz;
    int sh = hi * wsz + dh + (wsz >> 1); if (sh >= 128) sh -= 128;
    int sw = wi * wsz + dw + (wsz >> 1); if (sw >= 128) sw -= 128;
    size_t dst = (((size_t)b * 128 + sh) * 128 + sw) * 180 + chanOff + ci;
    ybuf[dst] = yp[idx];
}

// ---------------------------------------------------------------------------
// Host side
// ---------------------------------------------------------------------------
static inline void launch_gemm(const float* A, const float* B, const float* bias,
                               float* C, int M, int N, int K,
                               int lda, int ldb, int ldc, int transB,
                               hipStream_t stream)
{
    const int tm = M >> 7;               // M % 128 == 0 at every call site
    const int tn = (N + 63) >> 6;
    const int blocks = tm * tn;
    if (transB)
        gemm_f32_wmma<1><<<blocks, 256, 0, stream>>>(A, B, bias, C, M, N, K,
                                                     lda, ldb, ldc);
    else
        gemm_f32_wmma<0><<<blocks, 256, 0, stream>>>(A, B, bias, C, M, N, K,
                                                     lda, ldb, ldc);
}

extern "C" void kernel_launch(void* const* d_in, const int* in_sizes, int n_in,
                              void* d_out, int out_size, void* d_ws, size_t ws_size,
                              hipStream_t stream)
{
    const float* inp   = (const float*)d_in[0];  // (4,128,128,180)
    const float* w1    = (const float*)d_in[1];  // (1,1,180,360)
    const float* b1    = (const float*)d_in[2];  // (360)
    const float* gamma = (const float*)d_in[3];  // (360)
    const float* beta  = (const float*)d_in[4];  // (360)
    const float* w2    = (const float*)d_in[5];  // (1,1,180,180)
    const float* b2    = (const float*)d_in[6];  // (180)

    float* out = (float*)d_out;
    const int    M    = 65536;                   // B*H*W
    const size_t QVSZ = 3932160;                 // max N*D (= 4096*960)

    float* xbuf  = (float*)d_ws;                 // 65536 x 360
    float* qbuf  = xbuf + (size_t)M * 360;
    float* vbuf  = qbuf + QVSZ;
    float* ypk   = vbuf + QVSZ;
    float* ybuf  = ypk  + QVSZ;                  // 65536 x 180 (concat)
    float* stats = ybuf + (size_t)M * 180;       // 720 floats (sum, sumsq)

    // 1) conv1: (65536x180)*(180x360)+b1 -> xbuf
    launch_gemm(inp, w1, b1, xbuf, M, 360, 180, 180, 360, 360, 0, stream);

    // 2) batchnorm (training-mode batch stats)
    init_zero_kernel<<<1, 256, 0, stream>>>(stats, 720);
    bn_stats_kernel<<<1024, 256, 0, stream>>>(xbuf, stats, 64);
    bn_apply_kernel<<<(int)(((size_t)M * 360) / 256), 256, 0, stream>>>(
        xbuf, stats, gamma, beta, 1.0f / 65536.0f);

    // 3) three attention scales; S matrices live directly in d_out
    float* attn0 = out + 11796480;
    float* attn1 = attn0 + 16777216;
    float* attn2 = attn1 + 1048576;
    float* attn[3] = { attn0, attn1, attn2 };

    for (int s = 0; s < 3; ++s) {
        const int  wsz = 4 << s;                 // 4, 8, 16
        const int  g   = 128 / wsz;
        const int  N_  = 4 * g * g;              // windows
        const int  D_  = wsz * wsz * 60;         // window feature dim
        const long tot = (long)N_ * D_;
        const int  pb  = (int)((tot + 255) / 256);

        pack_qv_kernel<<<pb, 256, 0, stream>>>(xbuf, qbuf, vbuf, wsz, 120 * s,
                                               tot, D_, g);
        // S = Q * Q^T   (B stored row-major (N_,D_), transB)
        launch_gemm(qbuf, qbuf, nullptr, attn[s], N_, N_, D_, D_, D_, N_, 1, stream);
        softmax_rows_kernel<<<N_, 256, 0, stream>>>(attn[s], N_);
        // Y = S * V
        launch_gemm(attn[s], vbuf, nullptr, ypk, N_, D_, N_, N_, D_, D_, 0, stream);
        scatter_y_kernel<<<pb, 256, 0, stream>>>(ypk, ybuf, wsz, 60 * s,
                                                 tot, D_, g);
    }

    // 4) conv2: (65536x180)*(180x180)+b2 -> y output
    launch_gemm(ybuf, w2, b2, out, M, 180, 180, 180, 180, 180, 0, stream);
}